// SchurLayer_66365834658040
// MI455X (gfx1250) — compile-verified
//
#include <hip/hip_runtime.h>

// Problem constants
#define NDIM 128   // n
#define HIDN 512   // hidden

typedef __attribute__((ext_vector_type(2))) float v2f;
typedef __attribute__((ext_vector_type(8))) float v8f;

// Workspace layout (float offsets). Total = 4096 + 3*131072 = 397312 floats (~1.6 MB)
#define WS_COEF    0                    // 8 floats: alpha, beta, CU, CV, CK0, CD, CDBAR
#define WS_TOTAL   16                   // [2][512]
#define WS_DSUM    1040                 // [2][512]
#define WS_ROWSUM  4096                 // [2][128][512]  (becomes u' after finalize)
#define WS_COLSUM  (4096 + 131072)      // [2][128][512]  (becomes v  after finalize)
#define WS_DIAG    (4096 + 2*131072)    // [2][128][512]  (becomes diag-out after finalize)

// ---------------------------------------------------------------------------
// Coefficients: c_j = sum_i weight[i,j]*iso[i,j]; derive the 7 fused scalars.
// ---------------------------------------------------------------------------
__global__ void coef_kernel(const float* __restrict__ w,
                            const float* __restrict__ iso,
                            float* __restrict__ ws) {
    if (threadIdx.x == 0) {
        float c[7];
        for (int j = 0; j < 7; ++j) {
            float s = 0.0f;
            for (int i = 0; i < 7; ++i) s += w[i * 7 + j] * iso[i * 7 + j];
            c[j] = s;
        }
        // After accounting for the two (0,3,2,1) transposes, coefficients act on
        // projections of Y = x[s,:,:,h] directly with c4<->c5 swapped roles.
        float alpha = 0.5f * (c[5] + c[6]);   // on Y[i,j]
        float beta  = 0.5f * (c[5] - c[6]);   // on Y[j,i]
        float* cf = ws + WS_COEF;
        cf[0] = alpha;
        cf[1] = beta;
        cf[2] = c[4] - alpha;          // CU  : * xr[i]
        cf[3] = c[3] - alpha;          // CV  : * yc[j]
        cf[4] = c[2] - alpha - beta;   // CK0 : * s_off
        cf[5] = c[1];                  // CD  : * d[i]
        cf[6] = c[0] - c[1];           // CDBAR: * dbar
    }
}

// ---------------------------------------------------------------------------
// WMMA ones-reduction helper: A = all-ones 16x4 f32. D = sum over the 4 loaded
// rows per column, accumulated into C. Layout-robust because all A rows are
// identical (K-permutation invariant) and all D rows are equal.
// ---------------------------------------------------------------------------
__device__ __forceinline__ v8f wmma_ones_acc(v2f b, v8f c) {
    v2f a; a.x = 1.0f; a.y = 1.0f;
    return __builtin_amdgcn_wmma_f32_16x16x4_f32(false, a, false, b,
                                                 (short)0, c, false, false);
}

// ---------------------------------------------------------------------------
// Row sums + diagonal. grid = 256 (s,i), block = 1024 (32 waves = 32 h-chunks).
// Wave handles 16 h's; loops j in steps of 4 feeding WMMA B (4 rows x 16 h).
// ---------------------------------------------------------------------------
__global__ void rowsum_kernel(const float* __restrict__ x, float* __restrict__ ws) {
    const int si   = blockIdx.x;            // s*128 + i
    const int i    = si & 127;
    const int t    = threadIdx.x;
    const int lane = t & 31;
    const int h0   = (t >> 5) * 16;         // wave id -> h chunk
    const int nn   = lane & 15;
    const int half = lane >> 4;

    const float* rowbase = x + (size_t)si * NDIM * HIDN;   // x[s,i,0,0]

    v8f c = {};
    for (int j0 = 0; j0 < NDIM; j0 += 4) {
        const int j = j0 + 2 * half;
        v2f b;
        b.x = rowbase[(size_t)j       * HIDN + h0 + nn];
        b.y = rowbase[(size_t)(j + 1) * HIDN + h0 + nn];
        c = wmma_ones_acc(b, c);
    }
    if (lane < 16) {
        const size_t o = (size_t)si * HIDN + h0 + lane;
        ws[WS_ROWSUM + o] = c[0];
        ws[WS_DIAG   + o] = rowbase[(size_t)i * HIDN + h0 + lane];
    }
}

// ---------------------------------------------------------------------------
// Column sums. grid = 256 (s,j), block = 1024.
// ---------------------------------------------------------------------------
__global__ void colsum_kernel(const float* __restrict__ x, float* __restrict__ ws) {
    const int sj   = blockIdx.x;            // s*128 + j
    const int s    = sj >> 7;
    const int j    = sj & 127;
    const int t    = threadIdx.x;
    const int lane = t & 31;
    const int h0   = (t >> 5) * 16;
    const int nn   = lane & 15;
    const int half = lane >> 4;

    const float* base = x + ((size_t)s * NDIM * NDIM + j) * HIDN;  // x[s,0,j,0]
    const size_t rstride = (size_t)NDIM * HIDN;

    v8f c = {};
    for (int i0 = 0; i0 < NDIM; i0 += 4) {
        const int i = i0 + 2 * half;
        v2f b;
        b.x = base[(size_t)i       * rstride + h0 + nn];
        b.y = base[(size_t)(i + 1) * rstride + h0 + nn];
        c = wmma_ones_acc(b, c);
    }
    if (lane < 16) {
        ws[WS_COLSUM + (size_t)sj * HIDN + h0 + lane] = c[0];
    }
}

// ---------------------------------------------------------------------------
// Per-(s,h) totals: total = sum_i rowsum, dsum = sum_i diag. grid = 2, block = 1024.
// ---------------------------------------------------------------------------
__global__ void totals_kernel(float* __restrict__ ws) {
    const int s    = blockIdx.x;
    const int t    = threadIdx.x;
    const int lane = t & 31;
    const int h0   = (t >> 5) * 16;
    const int nn   = lane & 15;
    const int half = lane >> 4;

    const float* rs = ws + WS_ROWSUM + (size_t)s * NDIM * HIDN;
    const float* dg = ws + WS_DIAG   + (size_t)s * NDIM * HIDN;

    v8f ct = {};
    v8f cd = {};
    for (int i0 = 0; i0 < NDIM; i0 += 4) {
        const int i = i0 + 2 * half;
        v2f b, bd;
        b.x  = rs[(size_t)i       * HIDN + h0 + nn];
        b.y  = rs[(size_t)(i + 1) * HIDN + h0 + nn];
        bd.x = dg[(size_t)i       * HIDN + h0 + nn];
        bd.y = dg[(size_t)(i + 1) * HIDN + h0 + nn];
        ct = wmma_ones_acc(b, ct);
        cd = wmma_ones_acc(bd, cd);
    }
    if (lane < 16) {
        ws[WS_TOTAL + (size_t)s * HIDN + h0 + lane] = ct[0];
        ws[WS_DSUM  + (size_t)s * HIDN + h0 + lane] = cd[0];
    }
}

// ---------------------------------------------------------------------------
// Finalize u', v, diag-out in place. grid = 256 (s,i), block = 512 (h).
// ---------------------------------------------------------------------------
__global__ void finalize_kernel(float* __restrict__ ws) {
    const int si = blockIdx.x;
    const int s  = si >> 7;
    const int h  = threadIdx.x;

    const float beta = ws[WS_COEF + 1];
    const float CU   = ws[WS_COEF + 2];
    const float CV   = ws[WS_COEF + 3];
    const float CK0  = ws[WS_COEF + 4];
    const float CD   = ws[WS_COEF + 5];
    const float CDB  = ws[WS_COEF + 6];

    const float total = ws[WS_TOTAL + (size_t)s * HIDN + h];
    const float dsum  = ws[WS_DSUM  + (size_t)s * HIDN + h];

    const float s_off = (total - dsum) * (1.0f / (128.0f * 127.0f));
    const float dbar  = dsum * (1.0f / 128.0f);

    const size_t o = (size_t)si * HIDN + h;
    const float rs = ws[WS_ROWSUM + o];
    const float cs = ws[WS_COLSUM + o];
    const float dd = ws[WS_DIAG   + o];

    const float r = rs - dd - 127.0f * s_off;   // Op row sum
    const float c = cs - dd - 127.0f * s_off;   // Op col sum
    const float inv = 1.0f / (128.0f * 126.0f); // n*(n-2)
    const float xr = (127.0f * r + c) * inv;
    const float yc = (127.0f * c + r) * inv;

    ws[WS_ROWSUM + o] = CU * xr - beta * yc + CK0 * s_off;  // u'[i] (K0 folded)
    ws[WS_COLSUM + o] = CV * yc - beta * xr;                // v[j]
    ws[WS_DIAG   + o] = CD * dd + CDB * dbar;               // out diag
}

// ---------------------------------------------------------------------------
// Bulk output: out[s,i,j,h] = alpha*Y[i,j] + beta*Y[j,i] + u'[i] + v[j] (i!=j),
// diag-out on i==j. grid = 32768 (s,i,j), block = 512 (h). Fully h-coalesced.
// ---------------------------------------------------------------------------
__global__ void output_kernel(const float* __restrict__ x,
                              const float* __restrict__ ws,
                              float* __restrict__ out) {
    const int idx = blockIdx.x;          // ((s*128)+i)*128 + j
    const int j = idx & 127;
    const int i = (idx >> 7) & 127;
    const int s = idx >> 14;
    const int h = threadIdx.x;

    const size_t base = (size_t)idx * HIDN;
    const int si = (s << 7) | i;
    const int sj = (s << 7) | j;

    if (i == j) {
        out[base + h] = ws[WS_DIAG + (size_t)si * HIDN + h];
        return;
    }
    const float alpha = ws[WS_COEF + 0];
    const float beta  = ws[WS_COEF + 1];

    const float a = x[base + h];                                     // Y[i,j]
    const float b = x[((size_t)sj * NDIM + i) * HIDN + h];           // Y[j,i]
    const float u = ws[WS_ROWSUM + (size_t)si * HIDN + h];
    const float v = ws[WS_COLSUM + (size_t)sj * HIDN + h];

    out[base + h] = alpha * a + beta * b + u + v;
}

// ---------------------------------------------------------------------------
extern "C" void kernel_launch(void* const* d_in, const int* in_sizes, int n_in,
                              void* d_out, int out_size, void* d_ws, size_t ws_size,
                              hipStream_t stream) {
    (void)in_sizes; (void)n_in; (void)out_size; (void)ws_size;
    const float* x   = (const float*)d_in[0];  // [2,128,128,512] f32
    const float* w   = (const float*)d_in[1];  // [7,7] f32
    const float* iso = (const float*)d_in[2];  // [7,7] f32
    float* out = (float*)d_out;
    float* ws  = (float*)d_ws;

    coef_kernel    <<<1,                 32,   0, stream>>>(w, iso, ws);
    rowsum_kernel  <<<2 * NDIM,          1024, 0, stream>>>(x, ws);
    colsum_kernel  <<<2 * NDIM,          1024, 0, stream>>>(x, ws);
    totals_kernel  <<<2,                 1024, 0, stream>>>(ws);
    finalize_kernel<<<2 * NDIM,          512,  0, stream>>>(ws);
    output_kernel  <<<2 * NDIM * NDIM,   512,  0, stream>>>(x, ws, out);
}